// VMHCPLoss_51075751084157
// MI455X (gfx1250) — compile-verified
//
#include <hip/hip_runtime.h>
#include <stdint.h>

// ---- problem constants (from reference setup_inputs) ----
#define NB     2
#define DD     160
#define HH     192
#define WW     224
#define WIN_R  4          // win//2
#define WSF    729.0f     // win^3
#define TS     16         // output tile (H and W)
#define HSZ    24         // TS + 2*WIN_R (halo tile)
#define DCHUNK 40
#define NCHUNK 4

#define FB     6          // flow N*C = 2*3
#define FD     80
#define FH     96
#define FW     112

typedef __attribute__((ext_vector_type(2))) float v2f;
typedef __attribute__((ext_vector_type(8))) float v8f;

// ---- CDNA5 async global->LDS (ASYNCcnt-tracked) ----
__device__ __forceinline__ void async_ld_f32(uint32_t lds_addr, const float* g) {
  asm volatile("global_load_async_to_lds_b32 %0, %1, off"
               :: "v"(lds_addr), "v"((unsigned long long)(uintptr_t)g)
               : "memory");
}
__device__ __forceinline__ void wait_async0() {
  asm volatile("s_wait_asynccnt 0" ::: "memory");
}
__device__ __forceinline__ uint32_t lds_lo(const void* p) {
  // generic (flat) pointer to LDS keeps the LDS byte offset in addr[31:0]
  return (uint32_t)(uintptr_t)p;
}

// ============================ NCC kernel ============================
__global__ void __launch_bounds__(256)
ncc_kernel(const float* __restrict__ I, const float* __restrict__ J,
           double* __restrict__ accum) {
  __shared__ float prod[5][HSZ*HSZ];   // f0=I, f1=J (async-loaded), f2=I*I, f3=J*J, f4=I*J
  __shared__ float xbuf[5][HSZ*TS];    // after x box-sum: [y 0..23][xo 0..15]
  __shared__ float s2[9][5][TS*TS];    // circular buffer of HW-summed planes

  const int tid   = threadIdx.x;
  const int wave  = tid >> 5;
  const int lane  = tid & 31;
  const int xbase = blockIdx.x * TS;
  const int ybase = blockIdx.y * TS;
  const int b     = blockIdx.z >> 2;   // batch
  const int chunk = blockIdx.z & 3;    // D chunk
  const int d0    = chunk * DCHUNK;

  // ---- precompute the band matrix chunks (A of Out = Band(16x24) x In(24x16)) ----
  // A f32 16x4 layout: lanes 0-15: M=0..15, K={4c,4c+1}; lanes 16-31: M=0..15, K={4c+2,4c+3}
  const int am  = lane & 15;
  const int akh = (lane >> 4) << 1;    // 0 or 2
  v2f bandA[6];
  #pragma unroll
  for (int c6 = 0; c6 < 6; ++c6) {
    const int k0 = 4*c6 + akh;
    bandA[c6].x = (k0   >= am && k0   <= am + 8) ? 1.f : 0.f;
    bandA[c6].y = (k0+1 >= am && k0+1 <= am + 8) ? 1.f : 0.f;
  }

  float acc = 0.f;

  for (int p = d0 - WIN_R; p < d0 + DCHUNK + WIN_R; ++p) {
    const int slot = ((p % 9) + 9) % 9;
    if (p < 0 || p >= DD) {
      // zero-padded plane outside the volume
      for (int idx = tid; idx < 5*TS*TS; idx += 256)
        (&s2[slot][0][0])[idx] = 0.f;
      __syncthreads();
    } else {
      // stage A: async-load the 24x24 halo plane of I and J straight into LDS
      const size_t base = ((size_t)b * DD + p) * (size_t)(HH*WW);
      for (int idx = tid; idx < HSZ*HSZ; idx += 256) {
        const int hy = idx / HSZ;
        const int hx = idx - hy*HSZ;
        const int gy = ybase + hy - WIN_R;
        const int gx = xbase + hx - WIN_R;
        if (gy >= 0 && gy < HH && gx >= 0 && gx < WW) {
          const size_t off = base + (size_t)gy*WW + (size_t)gx;
          async_ld_f32(lds_lo(&prod[0][idx]), I + off);
          async_ld_f32(lds_lo(&prod[1][idx]), J + off);
        } else {
          prod[0][idx] = 0.f;
          prod[1][idx] = 0.f;
        }
      }
      wait_async0();
      __syncthreads();
      // stage A2: pointwise products
      for (int idx = tid; idx < HSZ*HSZ; idx += 256) {
        const float iv = prod[0][idx];
        const float jv = prod[1][idx];
        prod[2][idx] = iv*iv;
        prod[3][idx] = jv*jv;
        prod[4][idx] = iv*jv;
      }
      __syncthreads();
      // stage B: box-sum along x (5 fields x 24 rows x 16 outputs)
      for (int idx = tid; idx < 5*HSZ*TS; idx += 256) {
        const int f = idx / (HSZ*TS);
        const int r = idx - f*(HSZ*TS);
        const int y  = r >> 4;
        const int xo = r & 15;
        const float* src = &prod[f][y*HSZ + xo];
        float s = 0.f;
        #pragma unroll
        for (int k = 0; k < 9; ++k) s += src[k];
        xbuf[f][y*TS + xo] = s;
      }
      __syncthreads();
      // stage C (WMMA): box-sum along y as Out(16x16) = Band(16x24) x In(24x16)
      // one wave per field; 6 chained V_WMMA_F32_16X16X4_F32 (K=24)
      if (wave < 5) {
        const int f = wave;
        const int n = lane & 15;
        // B f32 4x16 layout: lane = column n (mod 16); lanes 0-15: K={4c,4c+1},
        // lanes 16-31: K={4c+2,4c+3} (VGPR index walks K within the lane half)
        v8f cacc = {};
        #pragma unroll
        for (int c6 = 0; c6 < 6; ++c6) {
          const int k0 = 4*c6 + akh;
          v2f bv;
          bv.x = xbuf[f][ k0      * TS + n];
          bv.y = xbuf[f][(k0 + 1) * TS + n];
          cacc = __builtin_amdgcn_wmma_f32_16x16x4_f32(
              /*neg_a=*/false, bandA[c6], /*neg_b=*/false, bv,
              /*c_mod=*/(short)0, cacc, /*reuse_a=*/false, /*reuse_b=*/false);
        }
        // D layout: VGPR v -> row m = v + 8*(lane>>4), column n = lane&15
        const int half = lane >> 4;
        #pragma unroll
        for (int v = 0; v < 8; ++v)
          s2[slot][f][(v + 8*half)*TS + n] = cacc[v];
      }
      __syncthreads();
    }
    // emit output plane (p - WIN_R): box-sum along D = sum of the 9 slots
    if (p >= d0 + WIN_R) {
      float S0=0.f, S1=0.f, S2v=0.f, S3v=0.f, S4=0.f;
      #pragma unroll
      for (int sl = 0; sl < 9; ++sl) {
        S0  += s2[sl][0][tid];
        S1  += s2[sl][1][tid];
        S2v += s2[sl][2][tid];
        S3v += s2[sl][3][tid];
        S4  += s2[sl][4][tid];
      }
      const float inv = 1.f / WSF;
      const float uI = S0*inv, uJ = S1*inv;
      const float cross = S4  - uJ*S0 - uI*S1 + uI*uJ*WSF;
      const float Ivar  = S2v - 2.f*uI*S0 + uI*uI*WSF;
      const float Jvar  = S3v - 2.f*uJ*S1 + uJ*uJ*WSF;
      acc += cross*cross / (Ivar*Jvar + 1e-5f);
    }
    __syncthreads();   // protect s2 slot reuse against next iteration's writes
  }

  // block reduction (wave32) -> double atomic
  #pragma unroll
  for (int o = 16; o > 0; o >>= 1) acc += __shfl_down(acc, o, 32);
  float* wred = &xbuf[0][0];
  if (lane == 0) wred[wave] = acc;
  __syncthreads();
  if (wave == 0) {
    float v = (lane < 8) ? wred[lane] : 0.f;
    #pragma unroll
    for (int o = 4; o > 0; o >>= 1) v += __shfl_down(v, o, 32);
    if (lane == 0) atomicAdd(&accum[0], (double)v);
  }
}

// ============================ Grad kernel ============================
__global__ void __launch_bounds__(256)
grad_kernel(const float* __restrict__ F, double* __restrict__ accum) {
  const long long n = (long long)FB*FD*FH*FW;
  float sd = 0.f, sh = 0.f, sw = 0.f;
  for (long long idx = (long long)blockIdx.x*256 + threadIdx.x; idx < n;
       idx += (long long)gridDim.x*256) {
    const int iw = (int)(idx % FW);
    long long t = idx / FW;
    const int ih = (int)(t % FH);
    t /= FH;
    const int id = (int)(t % FD);
    const float v0 = F[idx];
    if (id < FD-1) { const float d = F[idx + (long long)FH*FW] - v0; sd += d*d; }
    if (ih < FH-1) { const float d = F[idx + FW] - v0;               sh += d*d; }
    if (iw < FW-1) { const float d = F[idx + 1] - v0;                sw += d*d; }
  }
  #pragma unroll
  for (int o = 16; o > 0; o >>= 1) {
    sd += __shfl_down(sd, o, 32);
    sh += __shfl_down(sh, o, 32);
    sw += __shfl_down(sw, o, 32);
  }
  __shared__ float r[3][8];
  const int wave = threadIdx.x >> 5, lane = threadIdx.x & 31;
  if (lane == 0) { r[0][wave] = sd; r[1][wave] = sh; r[2][wave] = sw; }
  __syncthreads();
  if (wave == 0) {
    float a = (lane < 8) ? r[0][lane] : 0.f;
    float bb = (lane < 8) ? r[1][lane] : 0.f;
    float c = (lane < 8) ? r[2][lane] : 0.f;
    #pragma unroll
    for (int o = 4; o > 0; o >>= 1) {
      a += __shfl_down(a, o, 32);
      bb += __shfl_down(bb, o, 32);
      c += __shfl_down(c, o, 32);
    }
    if (lane == 0) {
      atomicAdd(&accum[1], (double)a);   // d-axis diffs (reference dy)
      atomicAdd(&accum[2], (double)bb);  // h-axis diffs (reference dx)
      atomicAdd(&accum[3], (double)c);   // w-axis diffs (reference dz)
    }
  }
}

// ============================ Init / finalize ============================
__global__ void init_kernel(double* a) {
  if (threadIdx.x < 4) a[threadIdx.x] = 0.0;
}

__global__ void finalize_kernel(const double* __restrict__ a, float* __restrict__ out) {
  const double cc_cnt = (double)NB * DD * HH * WW;
  const double cd = (double)FB * (FD-1) * FH * FW;
  const double ch = (double)FB * FD * (FH-1) * FW;
  const double cw = (double)FB * FD * FH * (FW-1);
  const double sim  = -(a[0] / cc_cnt);
  const double grad = (a[1]/cd + a[2]/ch + a[3]/cw) / 3.0 * 2.0;  // * INT_DOWNSIZE
  out[0] = (float)(sim + 0.01 * grad);                            // + W_SM * smooth
}

// ============================ launch ============================
extern "C" void kernel_launch(void* const* d_in, const int* in_sizes, int n_in,
                              void* d_out, int out_size, void* d_ws, size_t ws_size,
                              hipStream_t stream) {
  const float* warped = (const float*)d_in[0];
  const float* T1w    = (const float*)d_in[1];
  const float* flow   = (const float*)d_in[2];
  double* accum = (double*)d_ws;     // [0]=cc sum, [1..3]=grad sq sums
  float* out = (float*)d_out;

  hipLaunchKernelGGL(init_kernel, dim3(1), dim3(32), 0, stream, accum);

  dim3 grid(WW/TS, HH/TS, NB*NCHUNK);    // 14 x 12 x 8 = 1344 blocks
  hipLaunchKernelGGL(ncc_kernel, grid, dim3(256), 0, stream, T1w, warped, accum);

  hipLaunchKernelGGL(grad_kernel, dim3(4096), dim3(256), 0, stream, flow, accum);

  hipLaunchKernelGGL(finalize_kernel, dim3(1), dim3(1), 0, stream, accum, out);
}